// ElectricOverflow_40157944217661
// MI455X (gfx1250) — compile-verified
//
#include <hip/hip_runtime.h>

// ---- problem constants (match reference) ----
#define NMV   1500000
#define NTRM  100000
#define NFIL  1500000
#define NTOT  (NMV + NTRM + NFIL)
#define NBX   1024
#define NBY   1024
#define NBINS (NBX * NBY)
#define TD    0.9f
#define KFIX  22
#define KMOV  4

typedef __attribute__((ext_vector_type(2))) float v2f;
typedef __attribute__((ext_vector_type(8))) float v8f;

// ---------------- zero the density map + accumulators ----------------
__global__ void eo_zero(float* __restrict__ dmap, float* __restrict__ acc) {
    int i = blockIdx.x * blockDim.x + threadIdx.x;
    if (i < NBINS) dmap[i] = 0.0f;
    if (i == 0) {
        acc[0] = 0.0f;                 // density cost sum
        ((unsigned*)acc)[1] = 0u;      // max bits (densities >= 0, uint order == float order)
    }
}

// ------------- fixed/terminal nodes: thread = (node, kx), loop ky -------------
__global__ void eo_scatter_fixed(const float* __restrict__ pos,
                                 const float* __restrict__ sxv,
                                 const float* __restrict__ syv,
                                 float* __restrict__ dmap) {
    int tid  = blockIdx.x * blockDim.x + threadIdx.x;
    int node = tid / KFIX;
    int kx   = tid - node * KFIX;
    if (node >= NTRM) return;
    int g = NMV + node;
    float x  = pos[g];
    float y  = pos[NTOT + g];
    float sx = sxv[g];
    float sy = syv[g];
    int bx0 = (int)floorf(x);
    int by0 = (int)floorf(y);

    int bx = bx0 + kx;
    if (bx < 0 || bx >= NBX) return;
    float lox = (float)bx;
    float ox  = fminf(x + sx, lox + 1.0f) - fmaxf(x, lox);
    if (ox <= 0.0f) return;
    float wx = ox * TD;                       // init map is pre-scaled by TD
    float* row = dmap + bx * NBY;

    #pragma unroll 1
    for (int ky = 0; ky < KFIX; ++ky) {
        int by = by0 + ky;
        if (by < 0 || by >= NBY) continue;
        float loy = (float)by;
        float oy  = fminf(y + sy, loy + 1.0f) - fmaxf(y, loy);
        if (oy > 0.0f) {
            __hip_atomic_fetch_add(row + by, wx * oy,
                                   __ATOMIC_RELAXED, __HIP_MEMORY_SCOPE_AGENT);
        }
    }
}

// ------------- movable + filler nodes: thread = node, 4x4 unrolled -------------
__global__ void eo_scatter_mov(const float* __restrict__ pos,
                               const float* __restrict__ sxv,
                               const float* __restrict__ syv,
                               float* __restrict__ dmap) {
    int i = blockIdx.x * blockDim.x + threadIdx.x;
    if (i >= NMV + NFIL) return;
    int g = (i < NMV) ? i : (i + NTRM);       // skip terminal block
    float x  = pos[g];
    float y  = pos[NTOT + g];
    float sx = sxv[g];
    float sy = syv[g];
    int bx0 = (int)floorf(x);
    int by0 = (int)floorf(y);

    float ox[KMOV], oy[KMOV];
    #pragma unroll
    for (int k = 0; k < KMOV; ++k) {
        int bx = bx0 + k;
        float lox = (float)bx;
        float o  = fminf(x + sx, lox + 1.0f) - fmaxf(x, lox);
        ox[k] = (bx >= 0 && bx < NBX) ? fmaxf(o, 0.0f) : 0.0f;
        int by = by0 + k;
        float loy = (float)by;
        float o2 = fminf(y + sy, loy + 1.0f) - fmaxf(y, loy);
        oy[k] = (by >= 0 && by < NBY) ? fmaxf(o2, 0.0f) : 0.0f;
    }
    #pragma unroll
    for (int kx = 0; kx < KMOV; ++kx) {
        if (ox[kx] <= 0.0f) continue;         // ox>0 implies bx in range
        float* row = dmap + (bx0 + kx) * NBY + by0;
        #pragma unroll
        for (int ky = 0; ky < KMOV; ++ky) {
            float w = ox[kx] * oy[ky];        // oy>0 implies by in range
            if (w > 0.0f) {
                __hip_atomic_fetch_add(row + ky, w,
                                       __ATOMIC_RELAXED, __HIP_MEMORY_SCOPE_AGENT);
            }
        }
    }
}

// ------------- reduce: cost = sum(relu(d - TD)), maxd = max(d) -------------
// Per-wave sum is folded with one V_WMMA_F32_16X16X4_F32:
//   A(16x4): lane L sets VGPR0 = partial (K=0 for L<16, K=2 for L>=16), VGPR1 = 0
//   B(4x16) = ones  ->  D[m][n] = partial[m] + partial[m+16]  (replicated over n)
//   lane0's 8 acc VGPRs = D[0..7][0], lane16's = D[8..15][0]; one shuffle + one atomic.
__global__ void eo_reduce(const float* __restrict__ dmap,
                          const unsigned char* __restrict__ mask,
                          float* __restrict__ acc) {
    int tid    = blockIdx.x * blockDim.x + threadIdx.x;
    int stride = gridDim.x * blockDim.x;
    float lsum = 0.0f;
    float lmax = 0.0f;
    for (int i = tid; i < NBINS; i += stride) {
        float d = dmap[i];
        if (mask[i]) d = TD;                  // padding_mask -> TD * bin_area (bin_area = 1)
        lsum += fmaxf(d - TD, 0.0f);
        lmax  = fmaxf(lmax, d);
    }

    // wave32 max reduction
    #pragma unroll
    for (int off = 16; off > 0; off >>= 1)
        lmax = fmaxf(lmax, __shfl_xor(lmax, off, 32));

    // wave32 sum reduction via WMMA (EXEC is all-ones: full blocks, no divergence here)
    v2f a; a.x = lsum; a.y = 0.0f;
    v2f b; b.x = 1.0f; b.y = 1.0f;
    v8f c = {};
    c = __builtin_amdgcn_wmma_f32_16x16x4_f32(
            /*neg_a=*/false, a, /*neg_b=*/false, b,
            /*c_mod=*/(short)0, c, /*reuse_a=*/false, /*reuse_b=*/false);
    float t = c[0] + c[1] + c[2] + c[3] + c[4] + c[5] + c[6] + c[7];
    float t_hi = __shfl(t, 16, 32);           // lane16's half-wave total

    int lane = threadIdx.x & 31;
    if (lane == 0) {
        __hip_atomic_fetch_add(&acc[0], t + t_hi,
                               __ATOMIC_RELAXED, __HIP_MEMORY_SCOPE_AGENT);
        atomicMax((unsigned*)&acc[1], __float_as_uint(lmax));
    }
}

// ------------- finalize: write the two outputs -------------
__global__ void eo_finalize(const float* __restrict__ acc, float* __restrict__ out) {
    if (blockIdx.x == 0 && threadIdx.x == 0) {
        out[0] = acc[0];                                   // density_cost
        out[1] = __uint_as_float(((const unsigned*)acc)[1]); // max_density / bin_area(=1)
    }
}

extern "C" void kernel_launch(void* const* d_in, const int* in_sizes, int n_in,
                              void* d_out, int out_size, void* d_ws, size_t ws_size,
                              hipStream_t stream) {
    const float*         pos  = (const float*)d_in[0];         // 2N
    const float*         sx   = (const float*)d_in[1];         // N
    const float*         sy   = (const float*)d_in[2];         // N
    // d_in[3], d_in[4]: bin centers (implied by XL=0, BSX=1 -> unused)
    const unsigned char* mask = (const unsigned char*)d_in[5]; // NBX*NBY bool

    float* dmap = (float*)d_ws;          // 1024*1024 floats = 4 MB
    float* acc  = dmap + NBINS;          // [sum, maxbits]
    float* out  = (float*)d_out;

    eo_zero<<<NBINS / 256, 256, 0, stream>>>(dmap, acc);

    int nfix_threads = NTRM * KFIX;
    eo_scatter_fixed<<<(nfix_threads + 255) / 256, 256, 0, stream>>>(pos, sx, sy, dmap);

    int nmov = NMV + NFIL;
    eo_scatter_mov<<<(nmov + 255) / 256, 256, 0, stream>>>(pos, sx, sy, dmap);

    eo_reduce<<<1024, 256, 0, stream>>>(dmap, mask, acc);

    eo_finalize<<<1, 32, 0, stream>>>(acc, out);
}